// AtomicNumberPooling_76974403879560
// MI455X (gfx1250) — compile-verified
//
#include <hip/hip_runtime.h>
#include <hip/hip_bf16.h>

// CDNA5 / gfx1250. wave32. WMMA shape 16x16x4 f32: A=16x4, B=4x16, C/D=16x16.
typedef __attribute__((ext_vector_type(2))) float v2f;
typedef __attribute__((ext_vector_type(8))) float v8f;

#define NUM_Z   100
#define G_SEG   512
#define D_FEAT  64
#define CH      16   // nodes staged per chunk (4 KB of f32 features)

// Async global->LDS path (ASYNCcnt). Guarded: falls back to sync copy if the
// toolchain doesn't declare the gfx1250 async builtins.
#if __has_builtin(__builtin_amdgcn_global_load_async_to_lds_b128) && \
    __has_builtin(__builtin_amdgcn_s_wait_asynccnt)
#define USE_ASYNC 1
#else
#define USE_ASYNC 0
#endif

// Builtin prototype (from hipcc diagnostic): first arg is a pointer to a
// 16-byte GNU vector in AS(1) (printed as "__device__"), second in AS(3).
typedef int vint4_vs __attribute__((vector_size(16)));
typedef vint4_vs __attribute__((address_space(1)))* gbl_v4p;
typedef vint4_vs __attribute__((address_space(3)))* lds_v4p;

// --- Kernel 1: segment boundaries from sorted batch ids ---------------------
// start[g] = first index i with batch[i] >= g ; start[G] = N
__global__ void seg_bounds_kernel(const int* __restrict__ batch,
                                  int* __restrict__ start, int N) {
  int i = blockIdx.x * blockDim.x + threadIdx.x;
  if (i >= N) return;
  int bi = batch[i];
  if (i == 0) {
    for (int g = 0; g <= bi; ++g) start[g] = 0;
  } else {
    int bp = batch[i - 1];
    for (int g = bp + 1; g <= bi; ++g) start[g] = i;
  }
  if (i == N - 1) {
    for (int g = bi + 1; g <= G_SEG; ++g) start[g] = N;
  }
}

// Stage CH nodes (features async -> LDS, z sync). Every wave issues exactly ONE
// b128 async instruction with full EXEC (tail addresses are clamped in-bounds;
// garbage is finite and killed by the zero one-hot row), so ASYNCcnt per wave
// advances by exactly 1 per staged chunk.
__device__ __forceinline__ void stage_chunk(const float* __restrict__ feat,
                                            const int* __restrict__ zrv,
                                            float* __restrict__ bfeat,
                                            int* __restrict__ bz,
                                            int base, int e,
                                            int tid, int wave, int lane) {
  const int fofs = wave * 128 + lane * 4;      // float offset within 1024-float chunk
  int gn = base + (fofs >> 6);                 // node this 16B piece belongs to
  if (gn >= e) gn = e - 1;                     // clamp (e >= 1 when called)
  const float* gsrc = feat + (size_t)gn * D_FEAT + (fofs & 63);
  float* ldst = bfeat + fofs;
#if USE_ASYNC
  __builtin_amdgcn_global_load_async_to_lds_b128(
      (gbl_v4p)(uintptr_t)gsrc, (lds_v4p)(uintptr_t)ldst, 0, 0);
#else
  *(float4*)ldst = *(const float4*)gsrc;
#endif
  if (tid < CH) {
    const int zn = base + tid;
    bz[tid] = (zn < e) ? (zrv[zn] - 1) : -1;   // -1 never matches a bin
  }
}

// --- Kernel 2: per-graph binned mean pooling via one-hot WMMA ---------------
// Block = 1 graph, 256 threads = 8 waves. Waves 0..6 own bin tiles [16w,16w+16),
// each holds 4 accumulators covering features [0,64) -> full 112x64 in regs.
__global__ __launch_bounds__(256)
void pool_wmma_kernel(const float* __restrict__ feat,
                      const int*  __restrict__ zrv,
                      const int*  __restrict__ start,
                      float* __restrict__ outp) {
  __shared__ float lds_feat[2][CH * D_FEAT];   // double-buffered, 2 x 4 KB
  __shared__ int   lds_z[2][CH];

  const int g    = blockIdx.x;
  const int s    = start[g];
  const int e    = start[g + 1];
  const int cnt  = e - s;
  const float inv = 1.0f / (float)((cnt > 1) ? cnt : 1);

  const int tid  = threadIdx.x;
  const int wave = tid >> 5;
  const int lane = tid & 31;
  const int mlo  = lane & 15;          // M index (A) / N index (B,C)
  const int khi  = (lane >> 4) * 2;    // K base for this half-wave (ISA 7.12.2)

  v8f c0 = {}, c1 = {}, c2 = {}, c3 = {};

  const int nchunks = (cnt + CH - 1) / CH;
  if (nchunks > 0)
    stage_chunk(feat, zrv, lds_feat[0], lds_z[0], s, e, tid, wave, lane);

  for (int ch = 0; ch < nchunks; ++ch) {
    const int buf = ch & 1;
    if (ch + 1 < nchunks) {            // prefetch next chunk into other buffer
      stage_chunk(feat, zrv, lds_feat[buf ^ 1], lds_z[buf ^ 1],
                  s + (ch + 1) * CH, e, tid, wave, lane);
#if USE_ASYNC
      __builtin_amdgcn_s_wait_asynccnt(1);   // retire chunk ch, keep ch+1 in flight
#endif
    } else {
#if USE_ASYNC
      __builtin_amdgcn_s_wait_asynccnt(0);
#endif
    }
    __syncthreads();

    if (wave < 7) {                    // wave-uniform branch: EXEC all-ones
      const int binbase = wave * 16;
      const float* bf = lds_feat[buf];
      const int*   bz = lds_z[buf];
      #pragma unroll
      for (int k4 = 0; k4 < CH / 4; ++k4) {
        const int kk = k4 * 4 + khi;
        const int z0 = bz[kk];
        const int z1 = bz[kk + 1];
        // A (16x4 one-hot): V0 = K=kk, V1 = K=kk+1, row index = mlo
        v2f a;
        a.x = (z0 == binbase + mlo) ? 1.0f : 0.0f;
        a.y = (z1 == binbase + mlo) ? 1.0f : 0.0f;
        // B (4x16 features): V0 = row kk, V1 = row kk+1, col = mlo
        const float* r0 = &bf[kk * D_FEAT];
        const float* r1 = &bf[(kk + 1) * D_FEAT];
        v2f b0, b1, b2, b3;
        b0.x = r0[mlo];        b0.y = r1[mlo];
        b1.x = r0[16 + mlo];   b1.y = r1[16 + mlo];
        b2.x = r0[32 + mlo];   b2.y = r1[32 + mlo];
        b3.x = r0[48 + mlo];   b3.y = r1[48 + mlo];
        c0 = __builtin_amdgcn_wmma_f32_16x16x4_f32(false, a, false, b0, (short)0, c0, false, false);
        c1 = __builtin_amdgcn_wmma_f32_16x16x4_f32(false, a, false, b1, (short)0, c1, false, false);
        c2 = __builtin_amdgcn_wmma_f32_16x16x4_f32(false, a, false, b2, (short)0, c2, false, false);
        c3 = __builtin_amdgcn_wmma_f32_16x16x4_f32(false, a, false, b3, (short)0, c3, false, false);
      }
    }
    __syncthreads();                   // reads of buf done before it is refilled
  }

  // Write D: C layout = VGPR v holds M = v + 8*(lane>=16), N = mlo.
  if (wave < 7) {
    float* og = outp + (size_t)g * (NUM_Z * D_FEAT);
    const int mhi = (lane >> 4) * 8;
    #pragma unroll
    for (int v = 0; v < 8; ++v) {
      const int m = wave * 16 + mhi + v;
      if (m < NUM_Z) {
        og[(size_t)m * D_FEAT +      mlo] = c0[v] * inv;
        og[(size_t)m * D_FEAT + 16 + mlo] = c1[v] * inv;
        og[(size_t)m * D_FEAT + 32 + mlo] = c2[v] * inv;
        og[(size_t)m * D_FEAT + 48 + mlo] = c3[v] * inv;
      }
    }
  }
}

// --- Launcher ---------------------------------------------------------------
extern "C" void kernel_launch(void* const* d_in, const int* in_sizes, int n_in,
                              void* d_out, int out_size, void* d_ws, size_t ws_size,
                              hipStream_t stream) {
  const float* feat  = (const float*)d_in[0];   // [N, 64] f32
  const int*   zrv   = (const int*)d_in[1];     // [N] int32 in [1,100]
  const int*   batch = (const int*)d_in[2];     // [N] int32 sorted in [0,512)
  float*       out   = (float*)d_out;           // [512, 6400] f32
  int*         start = (int*)d_ws;              // 513 ints of scratch

  const int N = in_sizes[1];

  seg_bounds_kernel<<<(N + 255) / 256, 256, 0, stream>>>(batch, start, N);
  pool_wmma_kernel<<<G_SEG, 256, 0, stream>>>(feat, zrv, start, out);
}